// PatchMamba_34213709479979
// MI455X (gfx1250) — compile-verified
//
#include <hip/hip_runtime.h>
#include <hip/hip_bf16.h>
#include <math.h>

#define NLAYER 4
#define DMODEL 512
#define DINNER 1024
#define DSTATE 16
#define DTRANK 32
#define DCONV  4
#define BB     4
#define LL     512
#define BL     (BB * LL)   // 2048 rows in all GEMMs

typedef float v2f __attribute__((ext_vector_type(2)));
typedef float v8f __attribute__((ext_vector_type(8)));

__device__ __forceinline__ float sigmoidf_(float x) {
    return 1.0f / (1.0f + __expf(-x));
}

// C[m][n] = sum_k A[m][k] * W[n][k]  (+bias[n]) (softplus?) (+addend[m][n])
// Each wave computes a 32x64 tile = 2 (M) x 4 (N) WMMA subtiles with
// V_WMMA_F32_16X16X4_F32, K stepped by 4. Per K-step: 6 float2 loads feed
// 8 independent WMMAs (8 accumulator chains -> no D->A/B RAW hazards).
// All predicates are wave-uniform, so EXEC is all-ones at every WMMA.
// Requires: M % 32 == 0, N % 64 == 0, K % 4 == 0 (true for all call sites).
__global__ void mamba_gemm_wmma_f32(const float* __restrict__ A, int lda,
                                    const float* __restrict__ W, int ldw,
                                    float* __restrict__ C, int ldc,
                                    int M, int N, int K,
                                    const float* __restrict__ addend,
                                    const float* __restrict__ bias,
                                    int act /*1 = softplus*/) {
    const int lane   = threadIdx.x & 31;
    const int wave   = blockIdx.x * (blockDim.x >> 5) + (threadIdx.x >> 5);
    const int tilesN = N >> 6;                 // 64-wide N tiles
    const int tilesM = M >> 5;                 // 32-tall M tiles
    if (wave >= tilesM * tilesN) return;
    const int tm = wave / tilesN;
    const int tn = wave - tm * tilesN;
    const int m0 = tm << 5, n0 = tn << 6;
    const int r  = lane & 15;                  // row within A/W fragment
    const int hi = lane >> 4;                  // 0: K=0,1   1: K=2,3 (f32 A layout)

    const float* arow0 = A + (size_t)(m0 + r) * lda + 2 * hi;
    const float* arow1 = arow0 + (size_t)16 * lda;
    const float* wrow  = W + (size_t)(n0 + r) * ldw + 2 * hi;
    const size_t wstep = (size_t)16 * ldw;     // next 16-col N subtile

    v8f acc[2][4];
#pragma unroll
    for (int i = 0; i < 2; ++i)
#pragma unroll
        for (int j = 0; j < 4; ++j) acc[i][j] = (v8f){0.f,0.f,0.f,0.f,0.f,0.f,0.f,0.f};

    for (int k = 0; k < K; k += 4) {
        v2f a0 = *(const v2f*)(arow0 + k);
        v2f a1 = *(const v2f*)(arow1 + k);
        v2f b[4];
#pragma unroll
        for (int j = 0; j < 4; ++j) b[j] = *(const v2f*)(wrow + j * wstep + k);
#pragma unroll
        for (int j = 0; j < 4; ++j) {
            acc[0][j] = __builtin_amdgcn_wmma_f32_16x16x4_f32(
                false, a0, false, b[j], (short)0, acc[0][j], false, false);
            acc[1][j] = __builtin_amdgcn_wmma_f32_16x16x4_f32(
                false, a1, false, b[j], (short)0, acc[1][j], false, false);
        }
    }

    // D layout: VGPR i -> row base + i + 8*hi, col base + (lane&15)
#pragma unroll
    for (int sj = 0; sj < 4; ++sj) {
        const int col = n0 + sj * 16 + r;
        const float bv = bias ? bias[col] : 0.0f;
#pragma unroll
        for (int si = 0; si < 2; ++si) {
            const int rbase = m0 + si * 16 + (hi << 3);
#pragma unroll
            for (int i = 0; i < 8; ++i) {
                const int row = rbase + i;
                float v = acc[si][sj][i] + bv;
                if (act == 1) v = (v > 20.0f) ? v : log1pf(__expf(v));  // softplus
                if (addend) v += addend[(size_t)row * ldc + col];
                C[(size_t)row * ldc + col] = v;
            }
        }
    }
}

// Depthwise causal conv (D_CONV=4) + bias + SiLU.
// xc = XR[:, 0:DINNER] (XR is [BL][2*DINNER]); U is [BL][DINNER].
__global__ void mamba_conv_silu(const float* __restrict__ XR,
                                const float* __restrict__ cw,   // [DINNER][DCONV]
                                const float* __restrict__ cb,   // [DINNER]
                                float* __restrict__ U) {
    int idx = blockIdx.x * blockDim.x + threadIdx.x;
    if (idx >= BL * DINNER) return;
    int c  = idx & (DINNER - 1);
    int bl = idx >> 10;               // DINNER == 1024
    int l  = bl & (LL - 1);
    float s = cb[c];
    const float* w = cw + c * DCONV;
#pragma unroll
    for (int k = 0; k < DCONV; ++k) {
        int ls = l - (DCONV - 1) + k;
        if (ls >= 0)
            s += XR[(size_t)(bl - l + ls) * (2 * DINNER) + c] * w[k];
    }
    U[idx] = s * sigmoidf_(s);        // SiLU
}

// Selective scan: one thread per (b, d) channel; 16-wide state in registers.
// Also applies y = (scan + u*D) * silu(res).
__global__ void mamba_scan(const float* __restrict__ XDBL,   // [BL][DTRANK+2*DSTATE]
                           const float* __restrict__ DELTA,  // [BL][DINNER] (softplus'd)
                           const float* __restrict__ U,      // [BL][DINNER]
                           const float* __restrict__ XR,     // [BL][2*DINNER] (res part)
                           const float* __restrict__ alog,   // [DINNER][DSTATE]
                           const float* __restrict__ Dp,     // [DINNER]
                           float* __restrict__ Y) {          // [BL][DINNER]
    int t = blockIdx.x * blockDim.x + threadIdx.x;
    if (t >= BB * DINNER) return;
    int d = t & (DINNER - 1);
    int b = t >> 10;

    float An[DSTATE], h[DSTATE];
#pragma unroll
    for (int n = 0; n < DSTATE; ++n) {
        An[n] = -__expf(alog[d * DSTATE + n]);
        h[n]  = 0.0f;
    }
    const float Dd = Dp[d];

    for (int l = 0; l < LL; ++l) {
        const int bl = b * LL + l;
        const float delta = DELTA[(size_t)bl * DINNER + d];
        const float uu    = U[(size_t)bl * DINNER + d];
        const float du    = delta * uu;
        const float* bc   = XDBL + (size_t)bl * (DTRANK + 2 * DSTATE);
        float acc = 0.0f;
#pragma unroll
        for (int n = 0; n < DSTATE; ++n) {
            const float dA = __expf(delta * An[n]);
            h[n] = dA * h[n] + du * bc[DTRANK + n];           // B_t
            acc += h[n] * bc[DTRANK + DSTATE + n];            // C_t
        }
        const float res = XR[(size_t)bl * (2 * DINNER) + DINNER + d];
        Y[(size_t)bl * DINNER + d] = (acc + uu * Dd) * (res * sigmoidf_(res));
    }
}

static inline void launch_gemm(const float* A, int lda, const float* W, int ldw,
                               float* C, int ldc, int M, int N, int K,
                               const float* addend, const float* bias, int act,
                               hipStream_t stream) {
    int waves  = (M >> 5) * (N >> 6);      // 32x64 tile per wave
    int blocks = (waves + 7) / 8;          // 8 waves (256 threads) per block
    mamba_gemm_wmma_f32<<<blocks, 256, 0, stream>>>(A, lda, W, ldw, C, ldc,
                                                    M, N, K, addend, bias, act);
}

extern "C" void kernel_launch(void* const* d_in, const int* in_sizes, int n_in,
                              void* d_out, int out_size, void* d_ws, size_t ws_size,
                              hipStream_t stream) {
    const float* x          = (const float*)d_in[0];
    const float* in_proj_w  = (const float*)d_in[1];
    const float* conv_w     = (const float*)d_in[2];
    const float* conv_b     = (const float*)d_in[3];
    const float* x_proj_w   = (const float*)d_in[4];
    const float* dt_proj_w  = (const float*)d_in[5];
    const float* dt_proj_b  = (const float*)d_in[6];
    const float* A_log      = (const float*)d_in[7];
    const float* D_param    = (const float*)d_in[8];
    const float* out_proj_w = (const float*)d_in[9];
    float* out = (float*)d_out;

    float* ws = (float*)d_ws;
    float* XR    = ws;  ws += (size_t)BL * 2 * DINNER;            // 2048 x 2048
    float* U     = ws;  ws += (size_t)BL * DINNER;                // 2048 x 1024
    float* XDBL  = ws;  ws += (size_t)BL * (DTRANK + 2 * DSTATE); // 2048 x 64
    float* DELTA = ws;  ws += (size_t)BL * DINNER;                // 2048 x 1024
    float* Y     = ws;  ws += (size_t)BL * DINNER;                // 2048 x 1024
    float* xb0   = ws;  ws += (size_t)BL * DMODEL;                // 2048 x 512
    float* xb1   = ws;

    const float* xcur = x;
    for (int i = 0; i < NLAYER; ++i) {
        const float* wi   = in_proj_w  + (size_t)i * 2 * DINNER * DMODEL;
        const float* cw   = conv_w     + (size_t)i * DINNER * DCONV;
        const float* cb   = conv_b     + (size_t)i * DINNER;
        const float* wx   = x_proj_w   + (size_t)i * (DTRANK + 2 * DSTATE) * DINNER;
        const float* wdt  = dt_proj_w  + (size_t)i * DINNER * DTRANK;
        const float* bdt  = dt_proj_b  + (size_t)i * DINNER;
        const float* alog = A_log      + (size_t)i * DINNER * DSTATE;
        const float* Dp   = D_param    + (size_t)i * DINNER;
        const float* wo   = out_proj_w + (size_t)i * DMODEL * DINNER;
        float* xnext = (i == NLAYER - 1) ? out : ((i & 1) ? xb1 : xb0);

        // 1) XR = xcur @ wi^T          [2048 x 2048]
        launch_gemm(xcur, DMODEL, wi, DMODEL, XR, 2 * DINNER,
                    BL, 2 * DINNER, DMODEL, nullptr, nullptr, 0, stream);

        // 2) U = silu(causal_dwconv(XR[:, :DINNER]) + cb)
        mamba_conv_silu<<<(BL * DINNER + 255) / 256, 256, 0, stream>>>(XR, cw, cb, U);

        // 3) XDBL = U @ wx^T           [2048 x 64]
        launch_gemm(U, DINNER, wx, DINNER, XDBL, DTRANK + 2 * DSTATE,
                    BL, DTRANK + 2 * DSTATE, DINNER, nullptr, nullptr, 0, stream);

        // 4) DELTA = softplus(XDBL[:, :DTRANK] @ wdt^T + bdt)   [2048 x 1024]
        launch_gemm(XDBL, DTRANK + 2 * DSTATE, wdt, DTRANK, DELTA, DINNER,
                    BL, DINNER, DTRANK, nullptr, bdt, 1, stream);

        // 5) Y = selective_scan(...) * silu(res) + u*D gating
        mamba_scan<<<(BB * DINNER + 255) / 256, 256, 0, stream>>>(
            XDBL, DELTA, U, XR, alog, Dp, Y);

        // 6) xnext = xcur + Y @ wo^T   [2048 x 512]
        launch_gemm(Y, DINNER, wo, DINNER, xnext, DMODEL,
                    BL, DMODEL, DINNER, xcur, nullptr, 0, stream);

        xcur = xnext;
    }
}